// CorrelationGraphBuilder_14929306321606
// MI455X (gfx1250) — compile-verified
//
#include <hip/hip_runtime.h>
#include <math.h>
#include <stdint.h>

typedef __attribute__((ext_vector_type(2))) float v2f;
typedef __attribute__((ext_vector_type(4))) float v4f;
typedef __attribute__((ext_vector_type(8))) float v8f;

#define BATCH      128
#define NCH        64
#define TLEN       30720
#define TK         96                   // time-slab width staged in LDS
#define NT         (TLEN / TK)          // 320 slabs, exact
#define LDS_STRIDE 100                  // 96 + 4 pad -> conflict-free b64 reads
#define BUFSZ      (NCH * LDS_STRIDE)   // 6400 floats per buffer
#define PS_OFF     4096                 // partial row sums region (64*8 floats)
#define SUMS_OFF   4608                 // row sums (64)
#define NORMS_OFF  4672                 // row norms (64)

// Async DMA one 16B chunk: global (64-bit VA in VGPR pair) -> LDS (32-bit
// wave-relative offset in VGPR). Tracked by ASYNCcnt, no VGPR data round-trip.
__device__ __forceinline__ void async_copy16(const float* gsrc, float* ldst)
{
    const uint64_t ga = (uint64_t)(uintptr_t)gsrc;
    const uint32_t la = (uint32_t)(uintptr_t)ldst;   // low 32b = LDS offset
    asm volatile("global_load_async_to_lds_b128 %0, %1, off"
                 :: "v"(la), "v"(ga) : "memory");
}

#define WAIT_ASYNC(n) asm volatile("s_wait_asynccnt %0" :: "i"(n) : "memory")

// Issue one 64x96 fp32 slab (24,576 B) as async loads: 512 threads x 3 b128.
// Thread -> row = tid>>3, 8 threads cover a row (24 float4 => 3 per thread).
__device__ __forceinline__ void issue_slab(const float* __restrict__ xb, int t,
                                           float* lbuf, int tid)
{
    const int row = tid >> 3;
    const int c0  = tid & 7;
    #pragma unroll
    for (int k = 0; k < 3; ++k) {
        const int c = c0 + (k << 3);                 // float4 column 0..23
        async_copy16(xb + (size_t)row * TLEN + (size_t)t * TK + (c << 2),
                     lbuf + row * LDS_STRIDE + (c << 2));
    }
}

// One workgroup per batch. 512 threads = 16 wave32; each wave owns one 16x16
// tile of the 4x4 tile grid of the 64x64 Gram matrix.
__launch_bounds__(512)
__global__ void corr_gram_kernel(const float* __restrict__ x,
                                 float* __restrict__ adj)
{
    __shared__ __align__(16) float lds[2 * BUFSZ];   // 51,200 B, double buffer

    const int tid  = threadIdx.x;
    const int wave = tid >> 5;
    const int lane = tid & 31;
    const int b    = blockIdx.x;
    const float* xb = x + (size_t)b * NCH * TLEN;

    // Tile assignment: rows [ti, ti+16), cols [tj, tj+16)
    const int ti  = (wave & 3) << 4;
    const int tj  = (wave >> 2) << 4;
    const int sub = (lane >> 4) << 1;   // K sub-offset: 0 (lanes 0-15) / 2 (16-31)
    const int l15 = lane & 15;

    v8f acc = {};                       // 16x16 f32 accumulator (8 VGPRs)
    float rsum = 0.f;                   // partial sum of row (tid>>3)

    // Prologue: start DMA of slab 0 into buffer 0.
    issue_slab(xb, 0, lds, tid);

    for (int t = 0; t < NT; ++t) {
        float* cur = lds + (t & 1) * BUFSZ;

        if (t + 1 < NT) {
            // Start DMA of next slab into the other buffer (its previous
            // consumers finished before last iteration's trailing barrier),
            // then wait for THIS slab: in-order completion => <=3 outstanding.
            issue_slab(xb, t + 1, lds + ((t + 1) & 1) * BUFSZ, tid);
            WAIT_ASYNC(3);
        } else {
            WAIT_ASYNC(0);
        }
        __syncthreads();                // slab t visible workgroup-wide

        // Row-sum partials straight from LDS (each element counted once).
        {
            const int row = tid >> 3;
            const int c0  = tid & 7;
            #pragma unroll
            for (int k = 0; k < 3; ++k) {
                const int c = c0 + (k << 3);
                v4f q = *(const v4f*)&cur[row * LDS_STRIDE + (c << 2)];
                rsum += q.x + q.y + q.z + q.w;
            }
        }

        // 24 K-steps of V_WMMA_F32_16X16X4_F32 over the 64x96 slab.
        // A: 16x4 f32 layout -> lane L holds row (ti + L&15), K = k0+sub..+1.
        // B[k][n] = x[n][k]  -> same access pattern on column block tj.
        const float* Ap = cur + (ti + l15) * LDS_STRIDE + sub;
        const float* Bp = cur + (tj + l15) * LDS_STRIDE + sub;
        #pragma unroll
        for (int k0 = 0; k0 < TK; k0 += 4) {
            v2f a  = *(const v2f*)(Ap + k0);         // ds_load_b64
            v2f bt = *(const v2f*)(Bp + k0);         // ds_load_b64
            acc = __builtin_amdgcn_wmma_f32_16x16x4_f32(
                      false, a, false, bt, 0, acc, false, false);
        }
        __syncthreads();                // everyone done with slab t's buffer
    }

    // ---- Epilogue: Gram -> correlation -> threshold ----------------------
    // Dump accumulator into G[64][64] at LDS offset 0 (buffers are dead).
    // C/D layout: VGPR r -> M = r (lanes 0-15) or r+8 (lanes 16-31), N = lane&15.
    const int mrow = (lane >> 4) << 3;
    #pragma unroll
    for (int r = 0; r < 8; ++r)
        lds[(ti + mrow + r) * NCH + tj + l15] = acc[r];

    // Row-sum partials: 8 per row.
    lds[PS_OFF + (tid >> 3) * 8 + (tid & 7)] = rsum;
    __syncthreads();

    const float invT = 1.0f / (float)TLEN;
    if (tid < NCH) {
        float s = 0.f;
        #pragma unroll
        for (int i = 0; i < 8; ++i) s += lds[PS_OFF + tid * 8 + i];
        lds[SUMS_OFF + tid] = s;
        const float covnn = lds[tid * NCH + tid] - s * s * invT;
        lds[NORMS_OFF + tid] = sqrtf(fmaxf(covnn, 0.f));
    }
    __syncthreads();

    // corr = (G - s_n s_m / T) / (norm_n norm_m + eps); zero diag; threshold.
    float* adjb = adj + (size_t)b * NCH * NCH;
    for (int e = tid; e < NCH * NCH; e += 512) {
        const int n = e >> 6;
        const int m = e & 63;
        const float cov  = lds[e] - lds[SUMS_OFF + n] * lds[SUMS_OFF + m] * invT;
        float corr = cov / (lds[NORMS_OFF + n] * lds[NORMS_OFF + m] + 1e-12f);
        if (n == m) corr = 0.f;
        adjb[e] = (fabsf(corr) >= 0.5f) ? corr : 0.f;
    }
}

// Pass-through copy of x into d_out after adj. Pure 1 GB stream: non-temporal
// b128 load/store so we don't thrash L2 with data touched exactly once.
__global__ void copy_x_kernel(const v4f* __restrict__ src,
                              v4f* __restrict__ dst, long long n)
{
    long long i = (long long)blockIdx.x * blockDim.x + threadIdx.x;
    const long long stride = (long long)gridDim.x * blockDim.x;
    for (; i < n; i += stride) {
        v4f t = __builtin_nontemporal_load(src + i);
        __builtin_nontemporal_store(t, dst + i);
    }
}

extern "C" void kernel_launch(void* const* d_in, const int* in_sizes, int n_in,
                              void* d_out, int out_size, void* d_ws, size_t ws_size,
                              hipStream_t stream)
{
    const float* x = (const float*)d_in[0];
    float* out = (float*)d_out;

    // adj: first BATCH*NCH*NCH floats of d_out
    corr_gram_kernel<<<BATCH, 512, 0, stream>>>(x, out);

    // x pass-through: remaining BATCH*NCH*TLEN floats
    const long long nv4 = (long long)BATCH * NCH * TLEN / 4;  // 62,914,560 float4
    float* xout = out + (size_t)BATCH * NCH * NCH;
    copy_x_kernel<<<4096, 256, 0, stream>>>((const v4f*)x, (v4f*)xout, nv4);
}